// SbfLayer_19301583029077
// MI455X (gfx1250) — compile-verified
//
#include <hip/hip_runtime.h>
#include <math.h>

#define LMAX 7
#define NRAD 6
#define KTOT 42   // LMAX * NRAD
#define VEC  4
#define BLK  256

// out[(l*NRAD+n)*E + e] = coef[l,n] * j_l(z[l,n]*d[e]/5) * P_l(cos(alpha[e]))
// coef[l,n] = sqrt(2/(125*j_{l+1}(z)^2)) * sqrt((2l+1)/(4*pi))
__global__ __launch_bounds__(BLK) void sbf_kernel(
    const float* __restrict__ dists,
    const float* __restrict__ alphas,
    const float* __restrict__ zroots,
    float* __restrict__ out,
    int E)
{
    __shared__ float s_z[KTOT];
    __shared__ float s_cf[KTOT];
    const int tid = threadIdx.x;

    // ---- CDNA5 async global->LDS broadcast of the 42 Bessel roots ----
    if (tid < KTOT) {
        unsigned lds_off =
            (unsigned)(size_t)(__attribute__((address_space(3))) float*)(&s_z[tid]);
        const float* gp = zroots + tid;
        asm volatile("global_load_async_to_lds_b32 %0, %1, off"
                     :: "v"(lds_off), "v"(gp)
                     : "memory");
    }
    // each wave waits on its own ASYNCcnt; waves with none outstanding pass
    asm volatile("s_wait_asynccnt 0" ::: "memory");
    __syncthreads();

    // ---- per-block combined normalization coefficients (42 lanes, once) ----
    if (tid < KTOT) {
        const int   l   = tid / NRAD;
        const float x   = s_z[tid];
        float s, c;
        sincosf(x, &s, &c);
        const float inv = 1.0f / x;
        float jm = s * inv;               // j_0(x)
        float jc = (s * inv - c) * inv;   // j_1(x)
        for (int k = 1; k <= l; ++k) {    // upward to j_{l+1}(x)
            const float jn = (2.0f * (float)k + 1.0f) * inv * jc - jm;
            jm = jc; jc = jn;
        }
        const float norm = sqrtf(2.0f / (125.0f * jc * jc));
        const float pref = sqrtf((2.0f * (float)l + 1.0f) / 12.566370614359172f);
        s_cf[tid] = norm * pref;
    }
    __syncthreads();

    const int e0 = (blockIdx.x * BLK + tid) * VEC;
    if (e0 >= E) return;
    const bool full = (e0 + VEC <= E);

    float dv[VEC], av[VEC];
    if (full) {
        const float4 t0 = *(const float4*)(dists  + e0);
        const float4 t1 = *(const float4*)(alphas + e0);
        dv[0]=t0.x; dv[1]=t0.y; dv[2]=t0.z; dv[3]=t0.w;
        av[0]=t1.x; av[1]=t1.y; av[2]=t1.z; av[3]=t1.w;
    } else {
        for (int j = 0; j < VEC; ++j) {
            const int e = e0 + j;
            dv[j] = (e < E) ? dists[e]  : 0.0f;
            av[j] = (e < E) ? alphas[e] : 0.0f;
        }
    }

    float r[VEC], ca[VEC], Pm1[VEC], Pc[VEC];
    #pragma unroll
    for (int j = 0; j < VEC; ++j) {
        r[j]   = dv[j] / 5.0f;     // d / C
        ca[j]  = cosf(av[j]);
        Pm1[j] = 0.0f;             // unused until l>=1
        Pc[j]  = 1.0f;             // P_0
    }

    for (int l = 0; l < LMAX; ++l) {
        for (int n = 0; n < NRAD; ++n) {
            const int   k  = l * NRAD + n;
            const float zz = s_z[k];
            const float cf = s_cf[k];
            float o[VEC];
            #pragma unroll
            for (int j = 0; j < VEC; ++j) {
                const float x     = zz * r[j];
                const bool  small = fabsf(x) < 1e-8f;
                const float xs    = small ? 1e-8f : x;
                float s, c;
                sincosf(xs, &s, &c);
                const float inv = 1.0f / xs;
                const float j0  = s * inv;
                float jl;
                if (l == 0) {
                    jl = small ? 1.0f : j0;
                } else {
                    float jm = j0;
                    float jc = (s * inv - c) * inv;   // j_1
                    for (int kk = 1; kk < l; ++kk) {  // upward to j_l
                        const float jn = (2.0f * (float)kk + 1.0f) * inv * jc - jm;
                        jm = jc; jc = jn;
                    }
                    jl = small ? 0.0f : jc;
                }
                o[j] = cf * jl * Pc[j];
            }
            if (full) {
                *(float4*)(out + (size_t)k * (size_t)E + e0) =
                    make_float4(o[0], o[1], o[2], o[3]);
            } else {
                for (int j = 0; j < VEC; ++j) {
                    const int e = e0 + j;
                    if (e < E) out[(size_t)k * (size_t)E + e] = o[j];
                }
            }
        }
        // Legendre advance: P_{l+1} = ((2l+1)*ca*P_l - l*P_{l-1}) / (l+1)
        const float a     = 2.0f * (float)l + 1.0f;
        const float b     = (float)l;
        const float invl1 = 1.0f / ((float)l + 1.0f);
        #pragma unroll
        for (int j = 0; j < VEC; ++j) {
            const float Pn = (a * ca[j] * Pc[j] - b * Pm1[j]) * invl1;
            Pm1[j] = Pc[j];
            Pc[j]  = Pn;
        }
    }
}

extern "C" void kernel_launch(void* const* d_in, const int* in_sizes, int n_in,
                              void* d_out, int out_size, void* d_ws, size_t ws_size,
                              hipStream_t stream) {
    (void)n_in; (void)out_size; (void)d_ws; (void)ws_size;
    const float* d     = (const float*)d_in[0];
    const float* alpha = (const float*)d_in[1];
    const float* z     = (const float*)d_in[2];
    float*       out   = (float*)d_out;
    const int E = in_sizes[0];
    const int nthreads = (E + VEC - 1) / VEC;
    const int blocks   = (nthreads + BLK - 1) / BLK;
    sbf_kernel<<<blocks, BLK, 0, stream>>>(d, alpha, z, out, E);
}